// Attention_86431921864842
// MI455X (gfx1250) — compile-verified
//
#include <hip/hip_runtime.h>
#include <hip/hip_bf16.h>

typedef float v2f __attribute__((ext_vector_type(2)));
typedef float v8f __attribute__((ext_vector_type(8)));

static __device__ __forceinline__ v8f wmma4(v2f a, v2f b, v8f c) {
  // D = A(16x4 f32) * B(4x16 f32) + C(16x16 f32)
  return __builtin_amdgcn_wmma_f32_16x16x4_f32(false, a, false, b, (short)0, c,
                                               false, false);
}

static __device__ __forceinline__ void lds_fence() {
  asm volatile("s_wait_dscnt 0" ::: "memory");
}

// CDNA5 async global->LDS copy (ASYNCcnt-tracked), 8 bytes per lane.
static __device__ __forceinline__ void async_copy_b64(unsigned lds_off,
                                                      const float* gsrc) {
  asm volatile("global_load_async_to_lds_b64 %0, %1, off"
               :: "v"(lds_off), "v"(gsrc)
               : "memory");
}
static __device__ __forceinline__ void wait_async16() {
  asm volatile("s_wait_asynccnt 0x10" ::: "memory");
}
static __device__ __forceinline__ void wait_async0() {
  asm volatile("s_wait_asynccnt 0x0" ::: "memory");
}

static __device__ __forceinline__ float half_max(float x) {
  x = fmaxf(x, __shfl_xor(x, 1, 32));
  x = fmaxf(x, __shfl_xor(x, 2, 32));
  x = fmaxf(x, __shfl_xor(x, 4, 32));
  x = fmaxf(x, __shfl_xor(x, 8, 32));
  return x;
}
static __device__ __forceinline__ float half_sum(float x) {
  x += __shfl_xor(x, 1, 32);
  x += __shfl_xor(x, 2, 32);
  x += __shfl_xor(x, 4, 32);
  x += __shfl_xor(x, 8, 32);
  return x;
}

// ---------------------------------------------------------------------------
// Generic D = A * B^T GEMM (A: rows x K row-major, B: N x K row-major).
// One wave computes a 16x64 output block.  K-chunked: 40 loads in flight,
// then 32 WMMAs, so waits amortize instead of stalling per fragment.
// ---------------------------------------------------------------------------
__global__ __launch_bounds__(128) void gemm_nt_wmma(const float* __restrict__ A,
                                                    const float* __restrict__ B,
                                                    float* __restrict__ D,
                                                    int N, int K) {
  const int lane = threadIdx.x & 31;
  const int wid  = blockIdx.x * 4 + (threadIdx.x >> 5);
  const int hi = lane >> 4;      // 0: K pair {0,1}; 1: K pair {2,3}
  const int cc = lane & 15;      // M for A-frag, N for B-frag/D
  const int groups = (N >> 4) >> 2;   // 64-col groups
  const int rt = wid / groups;
  const int cg = wid - rt * groups;

  const float* arow = A + (size_t)(rt * 16 + cc) * K + 2 * hi;
  const float* b0 = B + (size_t)(cg * 64 +  0 + cc) * K + 2 * hi;
  const float* b1 = B + (size_t)(cg * 64 + 16 + cc) * K + 2 * hi;
  const float* b2 = B + (size_t)(cg * 64 + 32 + cc) * K + 2 * hi;
  const float* b3 = B + (size_t)(cg * 64 + 48 + cc) * K + 2 * hi;

  v8f d0 = {}, d1 = {}, d2 = {}, d3 = {};
  for (int k0 = 0; k0 < K; k0 += 32) {
    v2f a[8], bb0[8], bb1[8], bb2[8], bb3[8];
#pragma unroll
    for (int j = 0; j < 8; ++j) {
      a[j]   = *(const v2f*)(arow + k0 + 4 * j);
      bb0[j] = *(const v2f*)(b0 + k0 + 4 * j);
      bb1[j] = *(const v2f*)(b1 + k0 + 4 * j);
      bb2[j] = *(const v2f*)(b2 + k0 + 4 * j);
      bb3[j] = *(const v2f*)(b3 + k0 + 4 * j);
    }
#pragma unroll
    for (int j = 0; j < 8; ++j) {
      d0 = wmma4(a[j], bb0[j], d0);
      d1 = wmma4(a[j], bb1[j], d1);
      d2 = wmma4(a[j], bb2[j], d2);
      d3 = wmma4(a[j], bb3[j], d3);
    }
  }
#pragma unroll
  for (int r = 0; r < 8; ++r) {
    float* drow = D + (size_t)(rt * 16 + r + 8 * hi) * N + cg * 64 + cc;
    drow[0] = d0[r]; drow[16] = d1[r]; drow[32] = d2[r]; drow[48] = d3[r];
  }
}

// ---------------------------------------------------------------------------
// RoPE + pack: qkv_raw (256 x 3072) -> q2 (bh,16,128) pre-scaled by 1/sqrt(128),
// k2_new (bh,16,128), v_new (bh,16,64).  offset = 4096.
// ---------------------------------------------------------------------------
__global__ __launch_bounds__(256) void rope_pack(const float* __restrict__ qkv,
                                                 float* __restrict__ q2,
                                                 float* __restrict__ k2n,
                                                 float* __restrict__ vn) {
  int tid = blockIdx.x * 256 + threadIdx.x;   // 131072 total
  int i = tid & 31;
  int s = (tid >> 5) & 15;
  int h = (tid >> 9) & 15;
  int b = tid >> 13;
  int bh = b * 16 + h;
  const float* raw = qkv + (size_t)(b * 16 + s) * 3072 + h * 192;
  float inv = __expf(-(float)i * 0.28782313662425576f);  // 10000^(-i/32)
  float ang = (float)(4096 + s) * inv;
  float sn = __sinf(ang), cs = __cosf(ang);
  const float isc = 0.08838834764831845f;  // 1/sqrt(2*D_HEAD)

  float* q2b = q2  + ((size_t)bh * 16 + s) * 128;
  float* k2b = k2n + ((size_t)bh * 16 + s) * 128;
  float* vb  = vn  + ((size_t)bh * 16 + s) * 64;

  float x1 = raw[i], x2 = raw[32 + i];
  q2b[i]      = x1 * isc;
  q2b[32 + i] = x2 * isc;
  q2b[64 + i] = (x1 * cs - x2 * sn) * isc;
  q2b[96 + i] = (x1 * sn + x2 * cs) * isc;

  x1 = raw[64 + i]; x2 = raw[96 + i];
  k2b[i]      = x1;
  k2b[32 + i] = x2;
  k2b[64 + i] = x1 * cs - x2 * sn;
  k2b[96 + i] = x1 * sn + x2 * cs;

  vb[i]      = raw[128 + i];
  vb[32 + i] = raw[160 + i];
}

// ---------------------------------------------------------------------------
// Flash attention: one workgroup per (b,h), 4 waves split the 257 key tiles.
//  - QK: 32 batched global_load_b64 fragments -> 32 WMMAs.
//  - V : double-buffered async global->LDS staging (ASYNCcnt), coalesced
//        256B row copies, padded stride 72 for conflict-free ds reads.
//  - P : 16x16 transpose via LDS into WMMA A-layout.
// ---------------------------------------------------------------------------
#define FA_WAVES 4
#define VSTRIDE 72
__global__ __launch_bounds__(128) void flash_attn(
    const float* __restrict__ ck, const float* __restrict__ cv,
    const float* __restrict__ cp, const unsigned char* __restrict__ mask,
    const float* __restrict__ q2, const float* __restrict__ k2n,
    const float* __restrict__ vn, float* __restrict__ val) {
  __shared__ __align__(16) float vstage[FA_WAVES][2][16 * VSTRIDE];
  __shared__ __align__(16) float pT[FA_WAVES][16 * 18];
  __shared__ __align__(16) float accL[FA_WAVES][16 * 64];
  __shared__ float mL[FA_WAVES][16];
  __shared__ float lL[FA_WAVES][16];

  const int bh = blockIdx.x;            // 0..255  (b*16+h)
  const int lane = threadIdx.x & 31;
  const int wv = threadIdx.x >> 5;
  const int hi = lane >> 4;
  const int cc = lane & 15;

  // q2 A-fragments: aq[j] covers K dims [4j, 4j+4)
  const float* qb = q2 + (size_t)bh * 16 * 128 + cc * 128 + 2 * hi;
  v2f aq[32];
#pragma unroll
  for (int j = 0; j < 32; ++j) aq[j] = *(const v2f*)(qb + 4 * j);

  v8f zero = {};
  v8f acc[4] = {zero, zero, zero, zero};
  float m_run[8], l_run[8];
#pragma unroll
  for (int r = 0; r < 8; ++r) { m_run[r] = -1e30f; l_run[r] = 0.0f; }

  const float* vnew = vn + (size_t)bh * 1024;
  // ---- prologue: stage first V tile into buffer 0
  {
    const float* vsrc = (wv < 256)
        ? cv + ((size_t)bh * 4096 + (size_t)wv * 16) * 64
        : vnew;
#pragma unroll
    for (int kr = 0; kr < 16; ++kr)
      async_copy_b64(
          (unsigned)(uintptr_t)&vstage[wv][0][kr * VSTRIDE + 2 * lane],
          vsrc + kr * 64 + 2 * lane);
  }

  int parity = 0;
  for (int t = wv; t < 257; t += FA_WAVES) {
    const float *kb, *pb;
    int ks;
    if (t < 256) {
      size_t off = ((size_t)bh * 4096 + (size_t)t * 16);
      kb = ck + off * 64; pb = cp + off * 64; ks = 64;
    } else {
      kb = k2n + (size_t)bh * 2048; pb = kb + 64; ks = 128;
    }

    // ---- stage next V tile into the other buffer (async, ASYNCcnt)
    const int tn = t + FA_WAVES;
    if (tn < 257) {
      const float* vsrc = (tn < 256)
          ? cv + ((size_t)bh * 4096 + (size_t)tn * 16) * 64
          : vnew;
      lds_fence();  // prior ds reads of that buffer must be done (WAR)
#pragma unroll
      for (int kr = 0; kr < 16; ++kr)
        async_copy_b64(
            (unsigned)(uintptr_t)&vstage[wv][parity ^ 1][kr * VSTRIDE + 2 * lane],
            vsrc + kr * 64 + 2 * lane);
      // prefetch current K tile rows while we are at it
      __builtin_prefetch(kb, 0, 1);
    }

    // ---- scores: S = q2 (16x128) * k2_tile^T (128x16); batched loads first
    v2f kf[16], pf[16];
#pragma unroll
    for (int j = 0; j < 16; ++j)
      kf[j] = *(const v2f*)(kb + cc * ks + 4 * j + 2 * hi);
#pragma unroll
    for (int j = 0; j < 16; ++j)
      pf[j] = *(const v2f*)(pb + cc * ks + 4 * j + 2 * hi);
    v8f s = zero;
#pragma unroll
    for (int j = 0; j < 16; ++j) s = wmma4(aq[j], kf[j], s);
#pragma unroll
    for (int j = 0; j < 16; ++j) s = wmma4(aq[16 + j], pf[j], s);

    // ---- mask + online softmax (rows = queries live in VGPR index)
    const unsigned char* mrow =
        mask + (size_t)(8 * hi) * 4112 + (size_t)t * 16 + cc;
    float sv[8];
#pragma unroll
    for (int r = 0; r < 8; ++r) {
      float e = s[r];
      if (mrow[(size_t)r * 4112]) e = -1e30f;
      sv[r] = e;
    }
#pragma unroll
    for (int r = 0; r < 8; ++r) {
      float tmax = half_max(sv[r]);
      float mnew = fmaxf(m_run[r], tmax);
      float fcor = __expf(m_run[r] - mnew);
      m_run[r] = mnew;
      l_run[r] *= fcor;
      acc[0][r] *= fcor; acc[1][r] *= fcor;
      acc[2][r] *= fcor; acc[3][r] *= fcor;
      float pv = __expf(sv[r] - mnew);
      l_run[r] += half_sum(pv);
      sv[r] = pv;
    }

    // ---- transpose P (D layout: lane=key) -> A layout (lane=query) via LDS
    lds_fence();
#pragma unroll
    for (int r = 0; r < 8; ++r) pT[wv][(r + 8 * hi) * 18 + cc] = sv[r];
    lds_fence();

    // ---- make sure current V tile landed in LDS (next tile may be in flight)
    if (tn < 257) wait_async16(); else wait_async0();

    // ---- O += P (16x16) * V (16x64), V from staged LDS
    const float* vb = &vstage[wv][parity][0];
#pragma unroll
    for (int jk = 0; jk < 4; ++jk) {
      v2f ap = *(const v2f*)(&pT[wv][cc * 18 + 4 * jk + 2 * hi]);
      const float* v0 = vb + (4 * jk + 2 * hi) * VSTRIDE;
#pragma unroll
      for (int dt = 0; dt < 4; ++dt) {
        v2f bv;
        bv.x = v0[dt * 16 + cc];
        bv.y = v0[VSTRIDE + dt * 16 + cc];
        acc[dt] = wmma4(ap, bv, acc[dt]);
      }
    }
    parity ^= 1;
  }

  // ---- publish partials and merge across the 4 waves
#pragma unroll
  for (int dt = 0; dt < 4; ++dt)
#pragma unroll
    for (int r = 0; r < 8; ++r)
      accL[wv][(r + 8 * hi) * 64 + dt * 16 + cc] = acc[dt][r];
  if (cc == 0) {
#pragma unroll
    for (int r = 0; r < 8; ++r) {
      mL[wv][r + 8 * hi] = m_run[r];
      lL[wv][r + 8 * hi] = l_run[r];
    }
  }
  __syncthreads();

  if (wv == 0) {
    const int b = bh >> 4, h = bh & 15;
    for (int e = lane; e < 1024; e += 32) {
      int m = e >> 6, d = e & 63;
      float mm = mL[0][m];
#pragma unroll
      for (int w = 1; w < FA_WAVES; ++w) mm = fmaxf(mm, mL[w][m]);
      float ls = 0.0f, as = 0.0f;
#pragma unroll
      for (int w = 0; w < FA_WAVES; ++w) {
        float f = __expf(mL[w][m] - mm);
        ls += f * lL[w][m];
        as += f * accL[w][m * 64 + d];
      }
      // val layout: row = b*16+q, col = h*64+d (matches transpose+reshape)
      val[((size_t)(b * 16 + m)) * 1024 + h * 64 + d] = as / ls;
    }
  }
}

// ---------------------------------------------------------------------------
extern "C" void kernel_launch(void* const* d_in, const int* in_sizes, int n_in,
                              void* d_out, int out_size, void* d_ws,
                              size_t ws_size, hipStream_t stream) {
  (void)in_sizes; (void)n_in; (void)out_size; (void)ws_size;
  const float* x    = (const float*)d_in[0];
  const float* wqkv = (const float*)d_in[1];
  const float* wo   = (const float*)d_in[2];
  const float* ck   = (const float*)d_in[3];
  const float* cv   = (const float*)d_in[4];
  const float* cp   = (const float*)d_in[5];
  const unsigned char* mask = (const unsigned char*)d_in[6];
  float* out = (float*)d_out;

  float* ws      = (float*)d_ws;
  float* qkv_raw = ws;                       // 256*3072
  float* q2      = qkv_raw + 256 * 3072;     // 256*16*128
  float* k2n     = q2 + 524288;              // 256*16*128
  float* vn      = k2n + 524288;             // 256*16*64
  float* val     = vn + 262144;              // 256*1024

  // 1) qkv = x @ w_qkv^T : (256 x 1024) x (3072 x 1024)^T
  gemm_nt_wmma<<<192, 128, 0, stream>>>(x, wqkv, qkv_raw, 3072, 1024);
  // 2) RoPE + pack q2/k2_new/v_new
  rope_pack<<<512, 256, 0, stream>>>(qkv_raw, q2, k2n, vn);
  // 3) flash attention over 4096 cached + 16 new keys, D=128
  flash_attn<<<256, 128, 0, stream>>>(ck, cv, cp, mask, q2, k2n, vn, val);
  // 4) out = val @ w_o^T : (256 x 1024) x (1024 x 1024)^T
  gemm_nt_wmma<<<64, 128, 0, stream>>>(val, wo, out, 1024, 1024);
}